// GCN_15350213116645
// MI455X (gfx1250) — compile-verified
//
#include <hip/hip_runtime.h>

typedef float v2f __attribute__((ext_vector_type(2)));
typedef float v8f __attribute__((ext_vector_type(8)));

#define N_NODES 100000
#define N_EDGES 600000
#define IN_DIM  128
#define HID_DIM 128
#define OUT_DIM 64

// ---------------- degree kernels ----------------
__global__ void deg_init_kernel(float* __restrict__ deg) {
    int i = blockIdx.x * blockDim.x + threadIdx.x;
    if (i < N_NODES) deg[i] = 1.0f;   // self-loop baked in
}

__global__ void deg_accum_kernel(float* __restrict__ deg, const int* __restrict__ dst) {
    int e = blockIdx.x * blockDim.x + threadIdx.x;
    if (e < N_EDGES) unsafeAtomicAdd(&deg[dst[e]], 1.0f);  // global_atomic_add_f32
}

__global__ void deg_rsqrt_kernel(float* __restrict__ deg) {
    int i = blockIdx.x * blockDim.x + threadIdx.x;
    if (i < N_NODES) deg[i] = rsqrtf(deg[i]);
}

// ---------------- zero kernel (float4 stores) ----------------
__global__ void zero_kernel(float* __restrict__ p, int n4) {
    int i = blockIdx.x * blockDim.x + threadIdx.x;
    if (i < n4) {
        float4 z = make_float4(0.f, 0.f, 0.f, 0.f);
        ((float4*)p)[i] = z;
    }
}

// ---------------- WMMA f32 GEMM: H[M,N] = A[M,K] @ W[K,N] ----------------
// Block = 8 waves sharing ONE column tile. The 128x16 W tile is staged in LDS
// transposed with stride 132 floats: LDS bank = (4*l + k + kh) mod 64, so the
// per-wave ds_load_b64 B-fragment reads are bank-conflict-free.
// Each wave computes one 16x16 output tile via K/4 v_wmma_f32_16x16x4_f32 ops.
// A fragment (16x4, 2 VGPR): lanes 0-15 hold row M=l, K=k,k+1; lanes 16-31 K=k+2,k+3.
// B fragment (4x16, 2 VGPR): lanes 0-15 hold col N=l, K=k,k+1; lanes 16-31 K=k+2,k+3.
// D (16x16 f32, 8 VGPR): VGPR r -> row r (lanes 0-15) / row 8+r (lanes 16-31), col = lane&15.
#define WT_STRIDE 132

__global__ __launch_bounds__(256) void gemm_wmma_f32(const float* __restrict__ A,
                                                     const float* __restrict__ W,
                                                     float* __restrict__ H,
                                                     int M, int K, int N) {
    __shared__ float wt[16 * WT_STRIDE];   // W tile, transposed [col][k], 8448 B

    const int tid      = threadIdx.x;
    const int lane     = tid & 31;
    const int wid      = tid >> 5;
    const int colTiles = N >> 4;
    const int rowTiles = M >> 4;
    const int ct       = blockIdx.x % colTiles;
    const int rg       = blockIdx.x / colTiles;
    const int col0     = ct << 4;

    // cooperative, coalesced load of W[:, col0:col0+16] into LDS (transposed)
    for (int idx = tid; idx < (K << 4); idx += 256) {
        int k = idx >> 4;
        int c = idx & 15;
        wt[c * WT_STRIDE + k] = W[(size_t)k * N + col0 + c];
    }
    __syncthreads();

    const int rt = rg * 8 + wid;
    if (rt >= rowTiles) return;            // whole wave exits: EXEC stays all-1s

    const int row0 = rt << 4;
    const int l    = lane & 15;
    const int kh   = (lane >> 4) << 1;     // 0 for lanes 0-15, 2 for lanes 16-31

    const float* Arow = A + (size_t)(row0 + l) * K + kh;   // A[row0+l][kh + k]
    const float* Brow = wt + l * WT_STRIDE + kh;           // wt[l][kh + k]

    v8f c = {0.f, 0.f, 0.f, 0.f, 0.f, 0.f, 0.f, 0.f};
    #pragma unroll 4
    for (int k = 0; k < K; k += 4) {
        v2f a = { Arow[k], Arow[k + 1] };                  // global_load_b64
        v2f b = *(const v2f*)(Brow + k);                   // ds_load_b64, conflict-free
        c = __builtin_amdgcn_wmma_f32_16x16x4_f32(
                /*neg_a=*/false, a, /*neg_b=*/false, b,
                /*c_mod=*/(short)0, c, /*reuse_a=*/false, /*reuse_b=*/false);
    }

    const int mbase = row0 + ((lane >> 4) << 3);           // +8 rows for upper lane half
    float* out = H + (size_t)mbase * N + (col0 + l);
    #pragma unroll
    for (int r = 0; r < 8; ++r)
        out[(size_t)r * N] = c[r];
}

// ---------------- edge scatter: agg[dst] += dinv[src]*dinv[dst]*h[src] ----------------
__global__ void edge_agg_kernel(const float* __restrict__ h,
                                const int* __restrict__ src,
                                const int* __restrict__ dst,
                                const float* __restrict__ dinv,
                                float* __restrict__ agg, int D) {
    int t = blockIdx.x * blockDim.x + threadIdx.x;
    int chunks = D >> 2;
    int total = N_EDGES * chunks;
    if (t >= total) return;
    int e = t / chunks;
    int c = t - e * chunks;
    int s = src[e], d = dst[e];
    float coef = dinv[s] * dinv[d];
    float4 v = *(const float4*)(h + (size_t)s * D + (c << 2));
    float* o = agg + (size_t)d * D + (c << 2);
    unsafeAtomicAdd(o + 0, coef * v.x);
    unsafeAtomicAdd(o + 1, coef * v.y);
    unsafeAtomicAdd(o + 2, coef * v.z);
    unsafeAtomicAdd(o + 3, coef * v.w);
}

// ---------------- fused self-loop + bias (+ ReLU) ----------------
__global__ void finish_layer_kernel(const float* __restrict__ agg,
                                    const float* __restrict__ h,
                                    const float* __restrict__ dinv,
                                    const float* __restrict__ bias,
                                    float* __restrict__ out, int D, int doRelu) {
    int t = blockIdx.x * blockDim.x + threadIdx.x;
    int chunks = D >> 2;
    int total = N_NODES * chunks;
    if (t >= total) return;
    int i = t / chunks;
    int c = t - i * chunks;
    float di = dinv[i];
    float d2 = di * di;
    size_t off = (size_t)i * D + (c << 2);
    float4 av = *(const float4*)(agg + off);
    float4 hv = *(const float4*)(h + off);
    float4 bv = *(const float4*)(bias + (c << 2));
    float4 o;
    o.x = av.x + d2 * hv.x + bv.x;
    o.y = av.y + d2 * hv.y + bv.y;
    o.z = av.z + d2 * hv.z + bv.z;
    o.w = av.w + d2 * hv.w + bv.w;
    if (doRelu) {
        o.x = fmaxf(o.x, 0.f); o.y = fmaxf(o.y, 0.f);
        o.z = fmaxf(o.z, 0.f); o.w = fmaxf(o.w, 0.f);
    }
    *(float4*)(out + off) = o;
}

static inline int cdiv(long long a, int b) { return (int)((a + b - 1) / b); }

extern "C" void kernel_launch(void* const* d_in, const int* in_sizes, int n_in,
                              void* d_out, int out_size, void* d_ws, size_t ws_size,
                              hipStream_t stream) {
    const float* x   = (const float*)d_in[0];
    const int*   ei  = (const int*)d_in[1];
    const float* W1  = (const float*)d_in[2];
    const float* b1  = (const float*)d_in[3];
    const float* W2  = (const float*)d_in[4];
    const float* b2  = (const float*)d_in[5];
    const float* W3  = (const float*)d_in[6];
    const float* b3  = (const float*)d_in[7];
    const int* src = ei;            // edge_index[0]
    const int* dst = ei + N_EDGES;  // edge_index[1]
    float* out = (float*)d_out;

    // workspace layout (floats): dinv | hbuf | abuf
    float* dinv = (float*)d_ws;
    float* hbuf = dinv + ((N_NODES + 255) & ~255);
    float* abuf = hbuf + (size_t)N_NODES * HID_DIM;

    const int B = 256;

    // degrees -> dinv
    deg_init_kernel<<<cdiv(N_NODES, B), B, 0, stream>>>(dinv);
    deg_accum_kernel<<<cdiv(N_EDGES, B), B, 0, stream>>>(dinv, dst);
    deg_rsqrt_kernel<<<cdiv(N_NODES, B), B, 0, stream>>>(dinv);

    const int rowTiles   = N_NODES / 16;          // 6250, exact
    const int rowGroups  = cdiv(rowTiles, 8);     // 8 row tiles per block
    const int blocks128  = rowGroups * (HID_DIM / 16);
    const int blocks64   = rowGroups * (OUT_DIM / 16);

    // ---- layer 1 ----
    gemm_wmma_f32<<<blocks128, B, 0, stream>>>(x, W1, hbuf, N_NODES, IN_DIM, HID_DIM);
    zero_kernel<<<cdiv((long long)N_NODES * HID_DIM / 4, B), B, 0, stream>>>(abuf, N_NODES * HID_DIM / 4);
    edge_agg_kernel<<<cdiv((long long)N_EDGES * (HID_DIM / 4), B), B, 0, stream>>>(hbuf, src, dst, dinv, abuf, HID_DIM);
    finish_layer_kernel<<<cdiv((long long)N_NODES * (HID_DIM / 4), B), B, 0, stream>>>(abuf, hbuf, dinv, b1, abuf, HID_DIM, 1);

    // ---- layer 2 ----
    gemm_wmma_f32<<<blocks128, B, 0, stream>>>(abuf, W2, hbuf, N_NODES, HID_DIM, HID_DIM);
    zero_kernel<<<cdiv((long long)N_NODES * HID_DIM / 4, B), B, 0, stream>>>(abuf, N_NODES * HID_DIM / 4);
    edge_agg_kernel<<<cdiv((long long)N_EDGES * (HID_DIM / 4), B), B, 0, stream>>>(hbuf, src, dst, dinv, abuf, HID_DIM);
    finish_layer_kernel<<<cdiv((long long)N_NODES * (HID_DIM / 4), B), B, 0, stream>>>(abuf, hbuf, dinv, b2, abuf, HID_DIM, 1);

    // ---- layer 3 (aggregate straight into d_out, D = 64, no ReLU) ----
    gemm_wmma_f32<<<blocks64, B, 0, stream>>>(abuf, W3, hbuf, N_NODES, HID_DIM, OUT_DIM);
    zero_kernel<<<cdiv((long long)N_NODES * OUT_DIM / 4, B), B, 0, stream>>>(out, N_NODES * OUT_DIM / 4);
    edge_agg_kernel<<<cdiv((long long)N_EDGES * (OUT_DIM / 4), B), B, 0, stream>>>(hbuf, src, dst, dinv, out, OUT_DIM);
    finish_layer_kernel<<<cdiv((long long)N_NODES * (OUT_DIM / 4), B), B, 0, stream>>>(out, hbuf, dinv, b3, out, OUT_DIM, 0);
}